// HellingerKantorovichDistance_77996606095837
// MI455X (gfx1250) — compile-verified
//
#include <hip/hip_runtime.h>

// ---------------------------------------------------------------------------
// Hellinger-Kantorovich (Sinkhorn) distance, batched, for MI455X (gfx1250).
//
// Per 16 batch columns (one wave32):
//   D[48x16] = K_pad[48x40] x V[40x16] via 3x10 V_WMMA_F32_16X16X4_F32.
// K is symmetric and generated analytically into A-fragments (registers);
// a/b/u/v live in registers; fragment relayout via one shfl_xor(16) + select.
// ---------------------------------------------------------------------------

typedef float v2f __attribute__((ext_vector_type(2)));
typedef float v8f __attribute__((ext_vector_type(8)));

static constexpr int   kNBins  = 40;
static constexpr float kZMin   = 1e-6f;
static constexpr float kZMax   = 2.0f;
static constexpr float kInvLam = 2.0f;   // 1 / LAMBDA_REG (0.5)
static constexpr float kEps    = 1e-4f;
static constexpr int   kIters  = 50;
static constexpr int   kMT     = 3;      // M tiles: 48 rows (40 real + 8 pad)
static constexpr int   kKS     = 10;     // K steps: 40 / 4 (exact, no pad)

__device__ __forceinline__ float rcp_fast(float x) { return __builtin_amdgcn_rcpf(x); }
__device__ __forceinline__ float swap16(float x)   { return __shfl_xor(x, 16, 32); }

// linspace(Z_MIN, Z_MAX, 40) used by the cost matrix
__device__ __forceinline__ float cost_center(int j) {
  return kZMin + (float)j * ((kZMax - kZMin) / (float)(kNBins - 1));
}

// Convert a C/D-layout result (3 tiles of v8f rows) into B-fragments for the
// next 16x16x4 matmul.  For k-step s (rows 4s..4s+3) all four rows share one
// tile and one lane-half, so one cross-half swap + one select per register.
__device__ __forceinline__ void make_bfrags(const float dd[kMT][8], v2f out[kKS], bool lo) {
#pragma unroll
  for (int s = 0; s < kKS; ++s) {
    const int  r = 4 * s;
    const int  t = r >> 4;          // tile
    const int  q = r & 7;           // register index within tile
    const bool h = (r & 15) >= 8;   // which lane-half holds these rows in D
    const float r0 = dd[t][q + 0], r1 = dd[t][q + 1];
    const float r2 = dd[t][q + 2], r3 = dd[t][q + 3];
    float v0, v1;
    if (!h) {
      const float x2 = swap16(r2), x3 = swap16(r3);
      v0 = lo ? r0 : x2;
      v1 = lo ? r1 : x3;
    } else {
      const float x0 = swap16(r0), x1 = swap16(r1);
      v0 = lo ? x0 : r2;
      v1 = lo ? x1 : r3;
    }
    out[s].x = v0;
    out[s].y = v1;
  }
}

// Gaussian soft-histogram with the reference's double normalization, written
// directly into C/D fragment layout (row = 16t + p + 8*(lane>=16), col = lane&15).
__device__ __forceinline__ void build_dist(float x, float dd[kMT][8], bool lo) {
  const float hstep  = (kZMax - kZMin) / (float)kNBins;
  const float sigma  = 0.5f * hstep;
  const float inv2s2 = 1.0f / (2.0f * sigma * sigma);
  float local = 0.0f;
#pragma unroll
  for (int t = 0; t < kMT; ++t) {
#pragma unroll
    for (int p = 0; p < 8; ++p) {
      const int r = 16 * t + p + (lo ? 0 : 8);
      const float c = kZMin + ((float)r + 0.5f) * hstep;
      const float d = x - c;
      const float w = (r < kNBins) ? __expf(-d * d * inv2s2) : 0.0f;
      dd[t][p] = w;
      local += w;
    }
  }
  const float s1 = local + swap16(local);          // full 40-row sum
  const float s2 = s1 * rcp_fast(s1 + 1e-8f);      // sum after first normalize
  const float inv = rcp_fast((s1 + 1e-8f) * (s2 + 1e-8f));
#pragma unroll
  for (int t = 0; t < kMT; ++t)
#pragma unroll
    for (int p = 0; p < 8; ++p) dd[t][p] *= inv;
}

__global__ __launch_bounds__(256) void hk_sinkhorn_kernel(const float* __restrict__ pred,
                                                          const float* __restrict__ tgt,
                                                          float* __restrict__ partial) {
  const int  lane = threadIdx.x & 31;
  const int  wv   = threadIdx.x >> 5;
  const int  col  = lane & 15;
  const bool lo   = lane < 16;

  const long long base = ((long long)blockIdx.x * 8 + wv) * 16;
  const float xp = pred[base + col];
  const float xt = tgt[base + col];

  // ---- K in A-fragment layout (M = lane&15 per tile; k-cols split by half).
  // K symmetric => same fragments serve K and K^T.  Rows >= 40 zeroed (M pad).
  v2f kA[kMT][kKS];
#pragma unroll
  for (int t = 0; t < kMT; ++t) {
    const int   i  = 16 * t + col;
    const float ci = cost_center(i);
    const bool  ok = (i < kNBins);
#pragma unroll
    for (int s = 0; s < kKS; ++s) {
      const int   j0 = 4 * s + (lo ? 0 : 2);
      const float d0 = ci - cost_center(j0);
      const float d1 = ci - cost_center(j0 + 1);
      const float k0 = __expf(-(d0 * d0 + 1.0f) * kInvLam);
      const float k1 = __expf(-(d1 * d1 + 1.0f) * kInvLam);
      kA[t][s].x = ok ? k0 : 0.0f;
      kA[t][s].y = ok ? k1 : 0.0f;
    }
  }

  // ---- Marginals in D layout.
  float aD[kMT][8], bD[kMT][8];
  build_dist(xp, aD, lo);
  build_dist(xt, bD, lo);

  // ---- Sinkhorn iterations.
  v2f vB[kKS];  // v in B-fragment layout (rows 0..39 exactly; starts at ones)
#pragma unroll
  for (int s = 0; s < kKS; ++s) { vB[s].x = 1.0f; vB[s].y = 1.0f; }

  v2f   uB[kKS];
  float vD[kMT][8];

#pragma unroll 1
  for (int it = 0; it < kIters; ++it) {
    // Kv = K * v
    float uD[kMT][8];
#pragma unroll
    for (int t = 0; t < kMT; ++t) {
      v8f c = {0.f, 0.f, 0.f, 0.f, 0.f, 0.f, 0.f, 0.f};
#pragma unroll
      for (int s = 0; s < kKS; ++s)
        c = __builtin_amdgcn_wmma_f32_16x16x4_f32(false, kA[t][s], false, vB[s],
                                                  (short)0, c, false, false);
#pragma unroll
      for (int p = 0; p < 8; ++p)
        uD[t][p] = aD[t][p] * rcp_fast(c[p] + kEps);   // u = a / (Kv + eps)
    }
    make_bfrags(uD, uB, lo);

    // K^T u = K * u (symmetry)
#pragma unroll
    for (int t = 0; t < kMT; ++t) {
      v8f c = {0.f, 0.f, 0.f, 0.f, 0.f, 0.f, 0.f, 0.f};
#pragma unroll
      for (int s = 0; s < kKS; ++s)
        c = __builtin_amdgcn_wmma_f32_16x16x4_f32(false, kA[t][s], false, uB[s],
                                                  (short)0, c, false, false);
#pragma unroll
      for (int p = 0; p < 8; ++p)
        vD[t][p] = bD[t][p] * rcp_fast(c[p] + kEps);   // v = b / (K^T u + eps)
    }
    make_bfrags(vD, vB, lo);
  }

  // ---- hk^2 = sum_i v_i * [(K.C) u]_i   (symmetry of K.C), fragments on the fly.
  float part = 0.0f;
#pragma unroll
  for (int t = 0; t < kMT; ++t) {
    const int   i  = 16 * t + col;
    const float ci = cost_center(i);
    v8f c = {0.f, 0.f, 0.f, 0.f, 0.f, 0.f, 0.f, 0.f};
#pragma unroll
    for (int s = 0; s < kKS; ++s) {
      const int   j0 = 4 * s + (lo ? 0 : 2);
      const float d0 = ci - cost_center(j0);
      const float d1 = ci - cost_center(j0 + 1);
      v2f m2;
      m2.x = kA[t][s].x * (d0 * d0 + 1.0f);
      m2.y = kA[t][s].y * (d1 * d1 + 1.0f);
      c = __builtin_amdgcn_wmma_f32_16x16x4_f32(false, m2, false, uB[s],
                                                (short)0, c, false, false);
    }
#pragma unroll
    for (int p = 0; p < 8; ++p) part += vD[t][p] * c[p];
  }
  const float hk_sq = part + swap16(part);
  const float hk = __builtin_sqrtf(fmaxf(hk_sq, 0.0f) + 1e-8f);

  // ---- Deterministic reduction: wave -> block -> d_ws partials.
  float s = hk;  // duplicated across lane halves -> full-wave sum counts twice
#pragma unroll
  for (int off = 1; off < 32; off <<= 1) s += __shfl_xor(s, off, 32);
  const float waveSum = s * 0.5f;

  __shared__ float smem[8];
  if (lane == 0) smem[wv] = waveSum;
  __syncthreads();
  if (threadIdx.x == 0) {
    float b = 0.0f;
#pragma unroll
    for (int i = 0; i < 8; ++i) b += smem[i];
    partial[blockIdx.x] = b;
  }
}

__global__ __launch_bounds__(256) void hk_reduce_kernel(const float* __restrict__ partial,
                                                        float* __restrict__ out,
                                                        int nparts, float nbatch) {
  __shared__ float smem[256];
  float s = 0.0f;
  for (int i = threadIdx.x; i < nparts; i += 256) s += partial[i];
  smem[threadIdx.x] = s;
  __syncthreads();
  for (int off = 128; off > 0; off >>= 1) {
    if (threadIdx.x < off) smem[threadIdx.x] += smem[threadIdx.x + off];
    __syncthreads();
  }
  if (threadIdx.x == 0) out[0] = smem[0] / nbatch;
}

extern "C" void kernel_launch(void* const* d_in, const int* in_sizes, int n_in,
                              void* d_out, int out_size, void* d_ws, size_t ws_size,
                              hipStream_t stream) {
  const float* pred = (const float*)d_in[0];
  const float* tgt  = (const float*)d_in[1];
  float* out        = (float*)d_out;
  float* partial    = (float*)d_ws;

  const int n      = in_sizes[0];      // 1<<20
  const int blocks = n / 128;          // 8 waves * 16 batch cols per block

  hk_sinkhorn_kernel<<<blocks, 256, 0, stream>>>(pred, tgt, partial);
  hk_reduce_kernel<<<1, 256, 0, stream>>>(partial, out, blocks, (float)n);
}